// TSPEmbeddingNet_5660766896837
// MI455X (gfx1250) — compile-verified
//
#include <hip/hip_runtime.h>
#include <hip/hip_bf16.h>
#include <math.h>

// Problem constants (from reference): B=4, N=384, D=128, H=256
#define BATCH 4
#define NCITY 384
#define DIM   128
#define HID   256
#define BN    (BATCH * NCITY)   // 1536

typedef __attribute__((ext_vector_type(2))) float v2f;
typedef __attribute__((ext_vector_type(8))) float v8f;

// ---------------------------------------------------------------------------
// CDNA5 async global->LDS staging (ASYNCcnt path), with safe fallback.
// Builtin signature (from hipcc diagnostic): param0 is
//   'int __attribute__((vector_size(16))) __device__ *'  (addrspace(1), non-const)
// ---------------------------------------------------------------------------
#if defined(__gfx1250__) && __has_builtin(__builtin_amdgcn_global_load_async_to_lds_b128) && \
    __has_builtin(__builtin_amdgcn_s_wait_asynccnt)
#define HAVE_ASYNC_LDS 1
#else
#define HAVE_ASYNC_LDS 0
#endif

#if HAVE_ASYNC_LDS
typedef int v4i_b128 __attribute__((vector_size(16)));
typedef __attribute__((address_space(1))) v4i_b128* gptr_b128;
typedef __attribute__((address_space(3))) v4i_b128* lptr_b128;
#endif

__device__ __forceinline__ void cp16_g2l(float* dst_lds, const float* src_g)
{
#if HAVE_ASYNC_LDS
    __builtin_amdgcn_global_load_async_to_lds_b128(
        (gptr_b128)src_g, (lptr_b128)dst_lds, /*offset=*/0, /*cpol=*/0);
#else
    *(float4*)dst_lds = *(const float4*)src_g;
#endif
}

__device__ __forceinline__ void cp_g2l_wait()
{
#if HAVE_ASYNC_LDS
    __builtin_amdgcn_s_wait_asynccnt(0);
#endif
}

// ---------------------------------------------------------------------------
// Kernel 1: h1 = relu(cities @ We1 + be1)   [BN, HID]   (K=2, pure VALU)
// ---------------------------------------------------------------------------
__global__ __launch_bounds__(256) void k_h1(const float* __restrict__ cities,
                                            const float* __restrict__ We1,
                                            const float* __restrict__ be1,
                                            float* __restrict__ h1)
{
    int idx = blockIdx.x * 256 + threadIdx.x;   // over BN*HID
    int m = idx >> 8;          // row   (HID == 256)
    int k = idx & 255;         // hidden index
    float c0 = cities[m * 2 + 0];
    float c1 = cities[m * 2 + 1];
    float v = fmaf(c0, We1[k], fmaf(c1, We1[HID + k], be1[k]));
    h1[idx] = fmaxf(v, 0.0f);
}

// ---------------------------------------------------------------------------
// Generic f32 WMMA GEMM: out[M x Ncols] = opt_relu(A[M x K] @ W[K x Ncols] + bias)
// All strides are compile-time so B loads / C stores use immediate offsets.
// grid.x = M/16 row tiles, grid.y = Ncols/128 col blocks.
// Block = 256 threads = 8 waves; each wave owns a 16x16 output tile and
// iterates K in steps of 4 with V_WMMA_F32_16X16X4_F32.
// A tile (16 x K) staged in LDS via async b128 copies; row stride K+4 dwords
// makes the ds_load_2addr_b64 A-fragment reads tile all 64 banks conflict-free
// and keeps rows 16B-aligned.
// ---------------------------------------------------------------------------
template <int K, int LDA, int LDW, int LDO, bool RELU>
__global__ __launch_bounds__(256) void k_wmma_gemm(const float* __restrict__ A,
                                                   const float* __restrict__ W,
                                                   const float* __restrict__ bias,
                                                   float* __restrict__ out)
{
    __shared__ __align__(16) float As[16][K + 4];

    const int tid     = threadIdx.x;
    const int rowbase = blockIdx.x * 16;
    const int colblk  = blockIdx.y * 128;

    // Async cooperative staging of the 16 x K A-tile (b128 per lane)
    constexpr int K4 = K / 4;
    for (int idx = tid; idx < 16 * K4; idx += 256) {
        int m = idx / K4;
        int k = (idx % K4) * 4;
        cp16_g2l(&As[m][k], &A[(rowbase + m) * LDA + k]);
    }
    cp_g2l_wait();
    __syncthreads();

    const int lane = tid & 31;
    const int wave = tid >> 5;
    const int half = lane >> 4;     // 0: lanes 0-15, 1: lanes 16-31
    const int l16  = lane & 15;
    const int col  = colblk + wave * 16 + l16;

    const float* Wp = W + col;      // per-lane base; kk*LDW folds into IOFFSET
    // Prefetch the whole B stream for this wave (global_prefetch_b8)
#pragma unroll
    for (int pk = 0; pk < K; pk += 32)
        __builtin_prefetch(&Wp[pk * LDW], 0, 3);

    float bv = bias ? bias[col] : 0.0f;
    v8f acc;
#pragma unroll
    for (int r = 0; r < 8; ++r) acc[r] = bv;

#pragma unroll 4
    for (int k0 = 0; k0 < K; k0 += 4) {
        const int kk = k0 + 2 * half;
        // A fragment: lane l16 holds row M=l16; VGPR0=K(kk), VGPR1=K(kk+1)
        v2f a = *(const v2f*)&As[l16][kk];
        // B fragment: lane l16 holds col N; VGPR0=K(kk), VGPR1=K(kk+1)
        v2f b;
        b.x = Wp[(kk + 0) * LDW];
        b.y = Wp[(kk + 1) * LDW];
        acc = __builtin_amdgcn_wmma_f32_16x16x4_f32(
            /*neg_a=*/false, a, /*neg_b=*/false, b,
            /*c_mod=*/(short)0, acc, /*reuse_a=*/false, /*reuse_b=*/false);
    }

    float* op = out + (rowbase + 8 * half) * LDO + col;   // r*LDO -> immediate
#pragma unroll
    for (int r = 0; r < 8; ++r) {
        float v = acc[r];
        if (RELU) v = fmaxf(v, 0.0f);
        op[r * LDO] = v;            // M = r (+8 for lanes 16-31)
    }
}

// ---------------------------------------------------------------------------
// Kernel: attention logit GEMV  s[m] = h2[m,:] . wa2 + ba2   (one wave per row)
// ---------------------------------------------------------------------------
__global__ __launch_bounds__(256) void k_gemv(const float* __restrict__ h2,
                                              const float* __restrict__ wa2,
                                              const float* __restrict__ ba2,
                                              float* __restrict__ s)
{
    int lane = threadIdx.x & 31;
    int wave = threadIdx.x >> 5;
    int m = blockIdx.x * 8 + wave;
    const float* row = h2 + m * HID;
    float acc = 0.0f;
#pragma unroll
    for (int i = 0; i < HID / 32; ++i)
        acc = fmaf(row[lane + 32 * i], wa2[lane + 32 * i], acc);
#pragma unroll
    for (int off = 16; off > 0; off >>= 1)
        acc += __shfl_xor(acc, off, 32);
    if (lane == 0) s[m] = acc + ba2[0];
}

// ---------------------------------------------------------------------------
// Kernel: softmax over N per batch -> att (first 1536 floats of d_out)
// ---------------------------------------------------------------------------
__global__ __launch_bounds__(256) void k_softmax(const float* __restrict__ s,
                                                 float* __restrict__ att)
{
    __shared__ float red[256];
    __shared__ float vals[NCITY];
    const int b = blockIdx.x, tid = threadIdx.x;
    const float* sb = s + b * NCITY;

    float mx = -3.4e38f;
    for (int n = tid; n < NCITY; n += 256) { float v = sb[n]; vals[n] = v; mx = fmaxf(mx, v); }
    red[tid] = mx; __syncthreads();
    for (int off = 128; off > 0; off >>= 1) {
        if (tid < off) red[tid] = fmaxf(red[tid], red[tid + off]);
        __syncthreads();
    }
    mx = red[0]; __syncthreads();

    float sum = 0.0f;
    for (int n = tid; n < NCITY; n += 256) { float e = __expf(vals[n] - mx); vals[n] = e; sum += e; }
    red[tid] = sum; __syncthreads();
    for (int off = 128; off > 0; off >>= 1) {
        if (tid < off) red[tid] += red[tid + off];
        __syncthreads();
    }
    float inv = 1.0f / red[0];
    for (int n = tid; n < NCITY; n += 256) att[b * NCITY + n] = vals[n] * inv;
}

// ---------------------------------------------------------------------------
// Kernel: pairwise decoder. p[b,i,j] = sigmoid(sum_k relu(Ai[i,k]+Aj'[j,k])*wd2[k] + bd2)
// (bd1 was folded into Aj'). ReLU couples (i,j,k) -> not a GEMM; VALU + LDS tiles.
// Tile: 16 i x 32 j. LDS rows padded to 260 dwords (260 % 64 == 4, 16B-aligned
// rows) so the per-lane float4 reads of Aj stripe across banks; Ai reads are
// wave-uniform broadcasts (lane group shares i). Tiles staged with async b128.
// ---------------------------------------------------------------------------
#define TI 16
#define TJ 32
#define PAD 260
__global__ __launch_bounds__(256) void k_pairwise(const float* __restrict__ Ai,
                                                  const float* __restrict__ Aj,
                                                  const float* __restrict__ wd2,
                                                  const float* __restrict__ bd2,
                                                  float* __restrict__ p)
{
    __shared__ __align__(16) float Ais[TI][PAD];
    __shared__ __align__(16) float Ajs[TJ][PAD];
    __shared__ __align__(16) float wv[HID];

    const int tid   = threadIdx.x;
    const int ibase = blockIdx.x * TI;
    const int jbase = blockIdx.y * TJ;
    const int b     = blockIdx.z;
    const int rowoff = b * NCITY;

    // HID/4 = 64 b128 chunks per row
    for (int idx = tid; idx < TI * 64; idx += 256)
        cp16_g2l(&Ais[idx >> 6][(idx & 63) * 4],
                 &Ai[(rowoff + ibase + (idx >> 6)) * HID + (idx & 63) * 4]);
    for (int idx = tid; idx < TJ * 64; idx += 256)
        cp16_g2l(&Ajs[idx >> 6][(idx & 63) * 4],
                 &Aj[(rowoff + jbase + (idx >> 6)) * HID + (idx & 63) * 4]);
    if (tid < 64)
        cp16_g2l(&wv[tid * 4], &wd2[tid * 4]);
    cp_g2l_wait();
    __syncthreads();

    const float bias2 = bd2[0];
    const int j  = tid & 31;        // lane -> j (strided LDS banks)
    const int ig = tid >> 5;        // wave -> i group (wave-uniform)

#pragma unroll
    for (int ii = 0; ii < TI / 8; ++ii) {
        const int i = ig + 8 * ii;
        float acc = 0.0f;
#pragma unroll 4
        for (int k = 0; k < HID; k += 4) {
            float4 av = *(const float4*)&Ais[i][k];   // broadcast within wave
            float4 jv = *(const float4*)&Ajs[j][k];
            float4 w  = *(const float4*)&wv[k];       // broadcast
            acc = fmaf(fmaxf(av.x + jv.x, 0.0f), w.x, acc);
            acc = fmaf(fmaxf(av.y + jv.y, 0.0f), w.y, acc);
            acc = fmaf(fmaxf(av.z + jv.z, 0.0f), w.z, acc);
            acc = fmaf(fmaxf(av.w + jv.w, 0.0f), w.w, acc);
        }
        const int gi = ibase + i;
        const int gj = jbase + j;
        float val = (gi == gj) ? 0.0f : 1.0f / (1.0f + __expf(-(acc + bias2)));
        p[((size_t)(b * NCITY + gi)) * NCITY + gj] = val;
    }
}

// ---------------------------------------------------------------------------
// Host launcher
// ---------------------------------------------------------------------------
extern "C" void kernel_launch(void* const* d_in, const int* in_sizes, int n_in,
                              void* d_out, int out_size, void* d_ws, size_t ws_size,
                              hipStream_t stream)
{
    const float* cities = (const float*)d_in[0];
    const float* We1    = (const float*)d_in[1];
    const float* be1    = (const float*)d_in[2];
    const float* We2    = (const float*)d_in[3];
    const float* be2    = (const float*)d_in[4];
    const float* Wa1    = (const float*)d_in[5];
    const float* ba1    = (const float*)d_in[6];
    const float* wa2    = (const float*)d_in[7];
    const float* ba2    = (const float*)d_in[8];
    const float* Wd1    = (const float*)d_in[9];
    const float* bd1    = (const float*)d_in[10];
    const float* wd2    = (const float*)d_in[11];
    const float* bd2    = (const float*)d_in[12];

    float* out = (float*)d_out;
    float* att = out;            // [B, N]
    float* p   = out + BN;       // [B, N, N]

    // Workspace layout (floats); regions reused once dead: ~3.94 MB total.
    float* ws  = (float*)d_ws;
    float* h1  = ws;                       // BN*HID = 393216
    float* emb = h1 + BN * HID;            // BN*DIM = 196608
    float* h2  = emb + BN * DIM;           // BN*HID = 393216
    float* s   = h2 + BN * HID;            // BN     = 1536
    float* Ai  = h1;                       // reuse h1 region (dead after emb)
    float* Aj  = h2;                       // reuse h2 region (dead after s)

    // 1) h1 = relu(cities @ We1 + be1)
    k_h1<<<BN * HID / 256, 256, 0, stream>>>(cities, We1, be1, h1);

    // 2) emb = h1 @ We2 + be2          (K=256, Ncols=128)
    k_wmma_gemm<HID, HID, DIM, DIM, false><<<dim3(BN / 16, DIM / 128), 256, 0, stream>>>(
        h1, We2, be2, emb);

    // 3) h2 = relu(emb @ Wa1 + ba1)    (K=128, Ncols=256)
    k_wmma_gemm<DIM, DIM, HID, HID, true><<<dim3(BN / 16, HID / 128), 256, 0, stream>>>(
        emb, Wa1, ba1, h2);

    // 4) s = h2 . wa2 + ba2 ; att = softmax_N(s)
    k_gemv<<<BN / 8, 256, 0, stream>>>(h2, wa2, ba2, s);
    k_softmax<<<BATCH, 256, 0, stream>>>(s, att);

    // 5) Ai = emb @ Wd1[:D]           (K=128, no bias)
    k_wmma_gemm<DIM, DIM, HID, HID, false><<<dim3(BN / 16, HID / 128), 256, 0, stream>>>(
        emb, Wd1, nullptr, Ai);
    //    Aj = emb @ Wd1[D:] + bd1     (bias folded in)
    k_wmma_gemm<DIM, DIM, HID, HID, false><<<dim3(BN / 16, HID / 128), 256, 0, stream>>>(
        emb, Wd1 + DIM * HID, bd1, Aj);

    // 6) pairwise decoder -> p (diagonal zeroed)
    k_pairwise<<<dim3(NCITY / TI, NCITY / TJ, BATCH), 256, 0, stream>>>(
        Ai, Aj, wd2, bd2, p);
}